// BatchifyTERM_17669495456110
// MI455X (gfx1250) — compile-verified
//
#include <hip/hip_runtime.h>

// ---------------------------------------------------------------------------
// BatchifyTERM: segment gather -> padded (B, T, P, D) tile with zero fill.
// Pure data movement (~320 MiB @ 23.3 TB/s => ~14us floor). No FLOPs => no
// WMMA; the CDNA5-native path is the async global<->LDS DMA engine
// (ASYNCcnt). One block per term; 32KiB LDS staging, 8 async b128 loads in
// flight per lane between drains, NT cache policy both directions.
// ---------------------------------------------------------------------------

#define B_ 16
#define T_ 128
#define D_ 256
#define L_ 8192
#define ROWV_ (D_ / 4)      // 64 x float4 per row
#define GBATCH_ 8           // row-groups staged per LDS batch (8 * 4KiB = 32KiB)

typedef float v4f __attribute__((ext_vector_type(4)));
typedef int   v4i __attribute__((ext_vector_type(4)));

#if __has_builtin(__builtin_amdgcn_global_load_async_to_lds_b128) && \
    __has_builtin(__builtin_amdgcn_global_store_async_from_lds_b128) && \
    __has_builtin(__builtin_amdgcn_s_wait_asynccnt)
#define USE_ASYNC 1
#else
#define USE_ASYNC 0
#endif

#define CPOL_NT 1   // TH[2:0]=1 (non-temporal): both streams are touch-once

// Uniform (per-block) exclusive prefix sum over this batch's term lengths.
// Runs once per term now (2048 blocks total) -> negligible SALU cost.
__device__ __forceinline__ int term_offset(const int* __restrict__ lrow, int t) {
    int off = 0;
    for (int i = 0; i < t; ++i) off += lrow[i];
    return off;
}

#if USE_ASYNC
// Builtin signature (verified by hipcc diagnostics): (v4i __device__* gaddr,
// v4i __shared__* lds, imm offset, imm cpol).
typedef __attribute__((address_space(1))) v4i gv4i_t;   // global int4
typedef __attribute__((address_space(3))) v4i lv4i_t;   // LDS int4

// One block per term. Batched async DMA: stage GBATCH_ row-groups (32KiB) in
// LDS with 8 async loads in flight per lane, drain, 8 async stores, drain.
__global__ void __launch_bounds__(256)
pad_terms_async(const float* __restrict__ x, const int* __restrict__ lens,
                float* __restrict__ out, int P) {
    __shared__ v4f lbuf[GBATCH_ * 256];   // 32 KiB

    const int term = blockIdx.x;          // b * T + t
    const int b    = term >> 7;           // T_ == 128
    const int t    = term & (T_ - 1);
    const int tid  = threadIdx.x;
    const int rloc = tid >> 6;            // row within a 4-row group
    const int seg  = tid & (ROWV_ - 1);   // float4 column within row

    const int* lrow = lens + b * T_;
    const int  off  = term_offset(lrow, t);
    const int  len  = lrow[t];

    const v4f* srcBase = (const v4f*)x + ((long)b * L_ + off) * ROWV_ + seg;
    v4f*       dstBase = (v4f*)out + ((long)term * P) * ROWV_ + seg;

    const int G = (P + 3) >> 2;           // total 4-row groups (24 for P=96)

    for (int g0 = 0; g0 < G; g0 += GBATCH_) {
        const int gend = (g0 + GBATCH_ < G) ? (g0 + GBATCH_) : G;

        // Phase 1: fill LDS — async loads for data rows, DS zeros for padding.
        for (int g = g0; g < gend; ++g) {
            const int p    = (g << 2) + rloc;
            v4f*      slot = &lbuf[((g - g0) << 8) + tid];
            if (p < len) {
                __builtin_amdgcn_global_load_async_to_lds_b128(
                    (gv4i_t*)(srcBase + (long)p * ROWV_), (lv4i_t*)slot, 0, CPOL_NT);
            } else if (p < P) {
                *slot = (v4f){0.f, 0.f, 0.f, 0.f};
            }
        }

        // Drain this wave's async loads (LDS writes) and DS zero-writes.
        // Slots are lane-private: no cross-wave barrier needed.
        __builtin_amdgcn_s_wait_asynccnt(0);
#if __has_builtin(__builtin_amdgcn_s_wait_dscnt)
        __builtin_amdgcn_s_wait_dscnt(0);
#else
        __syncthreads();
#endif

        // Phase 2: async stores LDS -> output.
        for (int g = g0; g < gend; ++g) {
            const int p = (g << 2) + rloc;
            if (p < P) {
                v4f* slot = &lbuf[((g - g0) << 8) + tid];
                __builtin_amdgcn_global_store_async_from_lds_b128(
                    (gv4i_t*)(dstBase + (long)p * ROWV_), (lv4i_t*)slot, 0, CPOL_NT);
            }
        }

        // Stores must finish before the next batch overwrites the LDS buffer.
        __builtin_amdgcn_s_wait_asynccnt(0);
    }
    // S_ENDPGM's implicit wait-idle covers anything still outstanding.
}
#endif  // USE_ASYNC

// Fallback: direct b128 copy, non-temporal both ways.
__global__ void __launch_bounds__(256)
pad_terms_direct(const float* __restrict__ x, const int* __restrict__ lens,
                 float* __restrict__ out, int P) {
    const int term = blockIdx.y;
    const int b    = term >> 7;
    const int t    = term & (T_ - 1);
    const int tid  = threadIdx.x;
    const int p    = blockIdx.x * 4 + (tid >> 6);
    const int seg  = tid & (ROWV_ - 1);
    if (p >= P) return;

    const int* lrow = lens + b * T_;
    const int  off  = term_offset(lrow, t);
    const int  len  = lrow[t];

    v4f* dst = (v4f*)out + ((long)term * P + p) * ROWV_ + seg;
    if (p < len) {
        const v4f* src = (const v4f*)x + ((long)b * L_ + off + p) * ROWV_ + seg;
        v4f v = __builtin_nontemporal_load(src);
        __builtin_nontemporal_store(v, dst);
    } else {
        v4f z = {0.f, 0.f, 0.f, 0.f};
        __builtin_nontemporal_store(z, dst);
    }
}

extern "C" void kernel_launch(void* const* d_in, const int* in_sizes, int n_in,
                              void* d_out, int out_size, void* d_ws, size_t ws_size,
                              hipStream_t stream) {
    (void)in_sizes; (void)n_in; (void)d_ws; (void)ws_size;
    const float* x    = (const float*)d_in[0];
    const int*   lens = (const int*)d_in[1];   // JAX default x64-off => int32
    float*       out  = (float*)d_out;

    const int P = out_size / (B_ * T_ * D_);   // padded length (96 for this setup)

#if USE_ASYNC
    pad_terms_async<<<dim3(B_ * T_), 256, 0, stream>>>(x, lens, out, P);
#else
    pad_terms_direct<<<dim3((P + 3) / 4, B_ * T_), 256, 0, stream>>>(x, lens, out, P);
#endif
}